// mgfn_81166291959902
// MI455X (gfx1250) — compile-verified
//
#include <hip/hip_runtime.h>
#include <hip/hip_bf16.h>

typedef __attribute__((ext_vector_type(2))) float v2f;
typedef __attribute__((ext_vector_type(4))) float v4f;
typedef __attribute__((ext_vector_type(8))) float v8f;

#define F 1024
#define T 256
#define NCROPS 10
#define BS_TOT 32
#define BS_HALF 16
#define KSEL 3
#define NROWS (BS_TOT * NCROPS * T)   // 81920 (bc*t rows)
#define WAVES_PER_BLOCK 4

// ---------------------------------------------------------------------------
// Kernel 1: fused LayerNorm + GEMV(fc_w) via WMMA + row magnitude.
// One wave (32 lanes) owns 16 consecutive rows (each row = F floats).
// Phase A: cooperative coalesced stats (mean/var) per row, wave32 shfl reduce.
// Phase B: WMMA f32 16x16x4 accumulation of 16 row-dots with fc_w in B col 0.
// ---------------------------------------------------------------------------
__global__ __launch_bounds__(128) void k_ln_score(
    const float* __restrict__ x, const float* __restrict__ gamma,
    const float* __restrict__ beta, const float* __restrict__ fcw,
    const float* __restrict__ fcb,
    float* __restrict__ wsScores, float* __restrict__ wsMag,
    float* __restrict__ wsMu, float* __restrict__ wsRsig)
{
    __shared__ float sg[F], sb[F], sw[F];
    const int tid = threadIdx.x;
    for (int i = tid; i < F; i += 128) { sg[i] = gamma[i]; sb[i] = beta[i]; sw[i] = fcw[i]; }
    __syncthreads();

    const int lane = tid & 31;
    const int wave = tid >> 5;
    const long long tile = (long long)blockIdx.x * WAVES_PER_BLOCK + wave;
    const long long rowBase = tile * 16;

    // ---- Phase A: per-row mean / rsqrt(var+eps), fully coalesced b128 loads
    float myMu = 0.0f, myRs = 1.0f;
    for (int r = 0; r < 16; ++r) {
        const v4f* xr = (const v4f*)(x + (rowBase + r) * F);
        float s = 0.0f, ss = 0.0f;
        #pragma unroll
        for (int j = 0; j < 8; ++j) {
            v4f v = xr[j * 32 + lane];
            s  += v.x + v.y + v.z + v.w;
            ss += v.x * v.x + v.y * v.y + v.z * v.z + v.w * v.w;
        }
        #pragma unroll
        for (int m = 16; m >= 1; m >>= 1) {
            s  += __shfl_xor(s,  m, 32);
            ss += __shfl_xor(ss, m, 32);
        }
        const float mu  = s * (1.0f / (float)F);
        const float var = ss * (1.0f / (float)F) - mu * mu;
        const float rs  = rsqrtf(var + 1e-5f);
        if (lane == 0) { wsMu[rowBase + r] = mu; wsRsig[rowBase + r] = rs; }
        if ((lane & 15) == r) { myMu = mu; myRs = rs; }
    }

    // ---- Phase B: normalize + WMMA GEMV + magnitude.
    // A-matrix 16x4 f32 layout: lane L -> M = L&15, K = (L>>4)*2 (+1 in 2nd VGPR)
    // B-matrix 4x16: only column N==0 nonzero -> lanes 0 and 16 carry fc_w.
    const int rowInTile = lane & 15;
    const int kOff = (lane >> 4) * 2;
    const float* xrow = x + (rowBase + rowInTile) * F;
    const bool laneN0 = ((lane & 15) == 0);

    v8f acc = {};
    float magsq = 0.0f;
    for (int kb = 0; kb < F; kb += 4) {
        const int cp = kb + kOff;
        v2f xv = *(const v2f*)(xrow + cp);      // recent tile -> WGP$/L2 hit
        v2f gv = *(const v2f*)(sg + cp);
        v2f bv = *(const v2f*)(sb + cp);
        v2f wv = *(const v2f*)(sw + cp);
        v2f a;
        a.x = (xv.x - myMu) * myRs * gv.x + bv.x;
        a.y = (xv.y - myMu) * myRs * gv.y + bv.y;
        magsq += a.x * a.x + a.y * a.y;
        v2f bm;
        bm.x = laneN0 ? wv.x : 0.0f;
        bm.y = laneN0 ? wv.y : 0.0f;
        acc = __builtin_amdgcn_wmma_f32_16x16x4_f32(
            false, a, false, bm, (short)0, acc, false, false);
    }

    // magnitude: lane L and L^16 cover disjoint halves of row L&15
    const float otherHalf = __shfl_xor(magsq, 16, 32);
    const float mag = sqrtf(magsq + otherHalf);
    if (lane < 16) wsMag[rowBase + lane] = mag;

    // D column N=0: lane 0 VGPR r = row r dot; lane 16 VGPR r = row r+8 dot
    if ((lane & 15) == 0) {
        const float fb = fcb[0];
        const long long base = rowBase + (long long)(lane >> 4) * 8;
        #pragma unroll
        for (int r = 0; r < 8; ++r) {
            const float z = acc[r] + fb;
            wsScores[base + r] = 1.0f / (1.0f + __expf(-z));
        }
    }
}

// ---------------------------------------------------------------------------
// Kernel 2: mean over NCROPS crops of scores and magnitudes -> (32, 256)
// ---------------------------------------------------------------------------
__global__ __launch_bounds__(256) void k_avg(
    const float* __restrict__ wsScores, const float* __restrict__ wsMag,
    float* __restrict__ outScores, float* __restrict__ wsMagAvg)
{
    const int idx = blockIdx.x * blockDim.x + threadIdx.x;
    if (idx >= BS_TOT * T) return;
    const int b = idx >> 8;
    const int t = idx & 255;
    float ssum = 0.0f, msum = 0.0f;
    #pragma unroll
    for (int c = 0; c < NCROPS; ++c) {
        const int row = b * NCROPS + c;
        ssum += wsScores[row * T + t];
        msum += wsMag[row * T + t];
    }
    outScores[idx] = ssum * (1.0f / (float)NCROPS);
    wsMagAvg[idx]  = msum * (1.0f / (float)NCROPS);
}

// ---------------------------------------------------------------------------
// Kernel 3: top-3 over T per batch row (ties -> lower index, like lax.top_k),
// plus score means at selected indices.
// ---------------------------------------------------------------------------
__global__ void k_topk(
    const float* __restrict__ wsMagAvg, const float* __restrict__ outScores,
    int* __restrict__ wsIdx, float* __restrict__ outAbnScore,
    float* __restrict__ outNorScore)
{
    const int b = threadIdx.x;
    if (b >= BS_TOT) return;
    int chosen[KSEL];
    for (int j = 0; j < KSEL; ++j) {
        float best = -3.4e38f;
        int bi = 0;
        for (int t = 0; t < T; ++t) {
            bool used = false;
            for (int p = 0; p < j; ++p) used |= (chosen[p] == t);
            if (used) continue;
            const float v = wsMagAvg[b * T + t];
            if (v > best) { best = v; bi = t; }
        }
        chosen[j] = bi;
        wsIdx[b * KSEL + j] = bi;
    }
    float sm = 0.0f;
    for (int j = 0; j < KSEL; ++j) sm += outScores[b * T + chosen[j]];
    sm *= (1.0f / (float)KSEL);
    if (b < BS_HALF) outNorScore[b] = sm;
    else             outAbnScore[b - BS_HALF] = sm;
}

// ---------------------------------------------------------------------------
// Kernel 4: gather selected rows, recomputing feats from x + stored mu/rsig
// (instead of materializing the 335 MB feats tensor).
// out[(c*16+b), j, :] = LN(x[bcRow, idx[b][j], :])
// ---------------------------------------------------------------------------
__global__ __launch_bounds__(256) void k_gather(
    const float* __restrict__ x, const float* __restrict__ gamma,
    const float* __restrict__ beta, const float* __restrict__ wsMu,
    const float* __restrict__ wsRsig, const int* __restrict__ wsIdx,
    float* __restrict__ outAbnFeat, float* __restrict__ outNorFeat)
{
    const int blk = blockIdx.x;             // 0..959
    const int g   = blk / 480;              // 0=normal, 1=abnormal
    const int rem = blk % 480;
    const int b   = rem / 30;
    const int r2  = rem % 30;
    const int c   = r2 / KSEL;
    const int j   = r2 % KSEL;

    const int magRow = g * BS_HALF + b;
    const int tsel = wsIdx[magRow * KSEL + j];
    const long long bc = (long long)(g * BS_HALF + b) * NCROPS + c;
    const long long srcRow = bc * T + tsel;

    const float mu = wsMu[srcRow];
    const float rs = wsRsig[srcRow];

    const v4f* xr = (const v4f*)(x + srcRow * F);
    const v4f* gr = (const v4f*)gamma;
    const v4f* br = (const v4f*)beta;
    float* outp = (g ? outAbnFeat : outNorFeat) +
                  ((long long)(c * BS_HALF + b) * KSEL + j) * F;

    const int tid = threadIdx.x;            // 256 threads * float4 = 1024
    v4f xv = xr[tid];
    v4f gv = gr[tid];
    v4f bv = br[tid];
    v4f o;
    o.x = (xv.x - mu) * rs * gv.x + bv.x;
    o.y = (xv.y - mu) * rs * gv.y + bv.y;
    o.z = (xv.z - mu) * rs * gv.z + bv.z;
    o.w = (xv.w - mu) * rs * gv.w + bv.w;
    ((v4f*)outp)[tid] = o;
}

// ---------------------------------------------------------------------------
// Output layout (floats, concatenated in return order):
//   [0,16)        score_abnormal (16,1)
//   [16,32)       score_normal   (16,1)
//   [32,491552)   abn_feamagnitude (160,3,1024)
//   [491552,983072) nor_feamagnitude (160,3,1024)
//   [983072,991264) scores (32,256,1)
// ---------------------------------------------------------------------------
extern "C" void kernel_launch(void* const* d_in, const int* in_sizes, int n_in,
                              void* d_out, int out_size, void* d_ws, size_t ws_size,
                              hipStream_t stream) {
    const float* x     = (const float*)d_in[0];
    const float* gamma = (const float*)d_in[1];
    const float* beta  = (const float*)d_in[2];
    const float* fcw   = (const float*)d_in[3];
    const float* fcb   = (const float*)d_in[4];

    float* out = (float*)d_out;
    float* outAbnScore = out;
    float* outNorScore = out + 16;
    float* outAbnFeat  = out + 32;
    float* outNorFeat  = out + 491552;
    float* outScores   = out + 983072;

    float* ws = (float*)d_ws;
    float* wsScores = ws;                    // 81920
    float* wsMag    = ws + 81920;            // 81920
    float* wsMu     = ws + 163840;           // 81920
    float* wsRsig   = ws + 245760;           // 81920
    float* wsMagAvg = ws + 327680;           // 8192
    int*   wsIdx    = (int*)(ws + 335872);   // 96 ints

    // 81920 rows / 16 rows-per-wave / 4 waves-per-block = 1280 blocks
    k_ln_score<<<1280, 128, 0, stream>>>(x, gamma, beta, fcw, fcb,
                                         wsScores, wsMag, wsMu, wsRsig);
    k_avg<<<32, 256, 0, stream>>>(wsScores, wsMag, outScores, wsMagAvg);
    k_topk<<<1, 32, 0, stream>>>(wsMagAvg, outScores, wsIdx,
                                 outAbnScore, outNorScore);
    k_gather<<<960, 256, 0, stream>>>(x, gamma, beta, wsMu, wsRsig, wsIdx,
                                      outAbnFeat, outNorFeat);
}